// ConvNN_2D_Attn_K_All_20435454394594
// MI455X (gfx1250) — compile-verified
//
#include <hip/hip_runtime.h>

typedef __attribute__((ext_vector_type(2))) float v2f;
typedef __attribute__((ext_vector_type(8))) float v8f;

#define NTOK 256
#define BATCH 256
#define KNNB 9

__device__ __forceinline__ v8f wmma4(v2f a, v2f b, v8f c) {
    // D = A(16x4,f32) * B(4x16,f32) + C(16x16,f32)
    return __builtin_amdgcn_wmma_f32_16x16x4_f32(false, a, false, b, (short)0, c, false, false);
}

// ---------------------------------------------------------------------------
// Y = A * B^T + bias [+ReLU].  A: (M,K) row-major, Bm: (N,K) row-major,
// Y: (M,N).  One wave computes a 32x32 tile (2x2 register blocking, 4 f32
// accumulators).  K multiple of 16, M multiple of 32.  NB=true masks the N
// edge (fc2, N=10) with a 0/1 multiplier -- no divergent loads, no EXEC games.
// grid = (ceil(N/32), M/32), block = 32.
// ---------------------------------------------------------------------------
template <bool RELU, bool NB>
__global__ __launch_bounds__(32) void gemm_at_bt(
    const float* __restrict__ A, const float* __restrict__ Bm,
    const float* __restrict__ bias, float* __restrict__ Y,
    int M, int N, int K)
{
    const int lane = threadIdx.x & 31;
    const int hl   = lane >> 4;    // selects K pair {0,1} vs {2,3} of each 4-step
    const int l15  = lane & 15;

    const int m0 = blockIdx.y * 32 + l15;
    const int m1 = m0 + 16;
    const int n0 = blockIdx.x * 32 + l15;
    const int n1 = n0 + 16;

    const float nm0 = (!NB || n0 < N) ? 1.f : 0.f;
    const float nm1 = (!NB || n1 < N) ? 1.f : 0.f;
    const int   nc0 = NB ? (n0 < N ? n0 : N - 1) : n0;   // clamped row (always valid)
    const int   nc1 = NB ? (n1 < N ? n1 : N - 1) : n1;

    const float* a0 = A  + (size_t)m0  * K;
    const float* a1 = A  + (size_t)m1  * K;
    const float* b0 = Bm + (size_t)nc0 * K;
    const float* b1 = Bm + (size_t)nc1 * K;

    v8f c00 = {0.f,0.f,0.f,0.f,0.f,0.f,0.f,0.f};
    v8f c01 = c00, c10 = c00, c11 = c00;

    for (int kk = 0; kk < K; kk += 16) {
#pragma unroll
        for (int u = 0; u < 4; ++u) {
            const int kb = kk + u * 4 + hl * 2;
            v2f av0, av1, bv0, bv1;
            av0.x = a0[kb]; av0.y = a0[kb + 1];
            av1.x = a1[kb]; av1.y = a1[kb + 1];
            bv0.x = b0[kb]; bv0.y = b0[kb + 1];
            bv1.x = b1[kb]; bv1.y = b1[kb + 1];
            if (NB) {
                bv0.x *= nm0; bv0.y *= nm0;
                bv1.x *= nm1; bv1.y *= nm1;
            }
            c00 = wmma4(av0, bv0, c00);
            c01 = wmma4(av0, bv1, c01);
            c10 = wmma4(av1, bv0, c10);
            c11 = wmma4(av1, bv1, c11);
        }
    }

    const int mb0 = blockIdx.y * 32;
    const int mb1 = mb0 + 16;
#pragma unroll
    for (int r = 0; r < 8; ++r) {
        const int ro0 = r + hl * 8;   // C layout: row = r, hi half-lanes offset by 8
        float v00 = c00[r], v01 = c01[r], v10 = c10[r], v11 = c11[r];
        if (!NB || n0 < N) {
            const float bv = bias[n0];
            float x0 = v00 + bv, x1 = v10 + bv;
            if (RELU) { x0 = fmaxf(x0, 0.f); x1 = fmaxf(x1, 0.f); }
            Y[(size_t)(mb0 + ro0) * N + n0] = x0;
            Y[(size_t)(mb1 + ro0) * N + n0] = x1;
        }
        if (!NB || n1 < N) {
            const float bv = bias[n1];
            float x0 = v01 + bv, x1 = v11 + bv;
            if (RELU) { x0 = fmaxf(x0, 0.f); x1 = fmaxf(x1, 0.f); }
            Y[(size_t)(mb0 + ro0) * N + n1] = x0;
            Y[(size_t)(mb1 + ro0) * N + n1] = x1;
        }
    }
}

// ---------------------------------------------------------------------------
// pixel_unshuffle(s=2): x (256,3,32,32) -> xu (256,12,256 tokens)
// ---------------------------------------------------------------------------
__global__ void unshuffle12(const float* __restrict__ x, float* __restrict__ xu)
{
    int idx = blockIdx.x * blockDim.x + threadIdx.x;
    if (idx >= BATCH * 12 * NTOK) return;
    int t  = idx & 255;
    int cc = (idx >> 8) % 12;
    int b  = idx / (12 * NTOK);
    int hh = t >> 4, ww = t & 15;
    int c = cc >> 2, i = (cc >> 1) & 1, j = cc & 1;
    xu[idx] = x[((size_t)(b * 3 + c) * 32 + (hh * 2 + i)) * 32 + (ww * 2 + j)];
}

// ---------------------------------------------------------------------------
// Fused sim + top-9 + softmax.  One thread per (b, n) row.
// q,k: (B, CC, 256).  attn: (B,256,9) fp32, topi: (B,256,9) int.
// ---------------------------------------------------------------------------
template <int CC>
__global__ __launch_bounds__(128) void topk_attn(
    const float* __restrict__ q, const float* __restrict__ k,
    float* __restrict__ attn, int* __restrict__ topi)
{
    int gid = blockIdx.x * blockDim.x + threadIdx.x;
    if (gid >= BATCH * NTOK) return;
    const int b = gid >> 8;
    const int n = gid & 255;
    const float scale = rsqrtf((float)CC);

    const float* qb = q + (size_t)b * CC * NTOK;
    const float* kb = k + (size_t)b * CC * NTOK;

    float qc[CC];
#pragma unroll
    for (int c = 0; c < CC; ++c) qc[c] = qb[c * NTOK + n];

    float bestv[KNNB];
    int   besti[KNNB];
#pragma unroll
    for (int t = 0; t < KNNB; ++t) { bestv[t] = -3.0e38f; besti[t] = 0; }

    for (int m = 0; m < NTOK; ++m) {
        float s = 0.f;
#pragma unroll
        for (int c = 0; c < CC; ++c) s += qc[c] * kb[c * NTOK + m];
        s *= scale;
        if (s > bestv[KNNB - 1]) {              // strict '>' keeps earliest index on ties
            int p = KNNB - 1;
#pragma unroll
            for (int t = KNNB - 2; t >= 0; --t) {
                if (s > bestv[t]) {
                    bestv[t + 1] = bestv[t];
                    besti[t + 1] = besti[t];
                    p = t;
                }
            }
            bestv[p] = s; besti[p] = m;
        }
    }

    const float mx = bestv[0];
    float e[KNNB], sum = 0.f;
#pragma unroll
    for (int t = 0; t < KNNB; ++t) { e[t] = __expf(bestv[t] - mx); sum += e[t]; }
    const float inv = 1.f / sum;
    const size_t o = (size_t)gid * KNNB;
#pragma unroll
    for (int t = 0; t < KNNB; ++t) { attn[o + t] = e[t] * inv; topi[o + t] = besti[t]; }
}

// ---------------------------------------------------------------------------
// Gathered neighbor conv as WMMA GEMM.
// Per block: (batch b, 16-token tile).  Stage G in LDS in paired-K-row layout
//   Gp[(kc>>1)*16 + t] = { G[kc_even][t], G[kc_odd][t] }  (one ds_load_b64 per
// B fragment), then Out(CO x 16) = Wc(CO x CC*9) @ G with f32 WMMA.
// v: (B,CC,256), wc: (CO,CC,9), out: (B,CO,256).  block = 128 (4 waves).
// ---------------------------------------------------------------------------
template <int CC, int CO>
__global__ __launch_bounds__(128) void conv_gather_wmma(
    const float* __restrict__ v, const float* __restrict__ attn,
    const int* __restrict__ topi, const float* __restrict__ wc,
    const float* __restrict__ bc, float* __restrict__ out)
{
    constexpr int KC = CC * KNNB;            // 108 or 576, both multiples of 4
    __shared__ float Gp[KC * 16];            // paired layout, same total size
    __shared__ float sAttn[16 * KNNB];
    __shared__ int   sIdx[16 * KNNB];

    const int b   = blockIdx.y;
    const int n0  = blockIdx.x * 16;
    const int tid = threadIdx.x;

    const size_t abase = ((size_t)b * NTOK + n0) * KNNB;
    for (int i = tid; i < 16 * KNNB; i += 128) {
        sAttn[i] = attn[abase + i];
        sIdx[i]  = topi[abase + i];
    }
    __syncthreads();

    const float* vb = v + (size_t)b * CC * NTOK;
    for (int e = tid; e < KC * 16; e += 128) {
        const int t  = e & 15;
        const int kc = e >> 4;               // kc = c*9 + kn matches wc (CO,CC,9) flatten
        const int c  = kc / KNNB;
        const int kn = kc % KNNB;
        const float val = sAttn[t * KNNB + kn] * vb[c * NTOK + sIdx[t * KNNB + kn]];
        Gp[((kc >> 1) * 16 + t) * 2 + (kc & 1)] = val;
    }
    __syncthreads();

    const int wave = tid >> 5;
    const int lane = tid & 31;
    const int hl   = lane >> 4;
    const int l15  = lane & 15;
    const v2f* Gp2 = (const v2f*)Gp;

    for (int tm = wave; tm < CO / 16; tm += 4) {
        const float* arow = wc + (size_t)(tm * 16 + l15) * KC;
        v8f acc = {0.f, 0.f, 0.f, 0.f, 0.f, 0.f, 0.f, 0.f};
#pragma unroll 4
        for (int kk = 0; kk < KC; kk += 4) {
            const int kb = kk + hl * 2;      // always even
            v2f a, bf;
            a.x = arow[kb];
            a.y = arow[kb + 1];
            bf  = Gp2[(kb >> 1) * 16 + l15];
            acc = wmma4(a, bf, acc);
        }
#pragma unroll
        for (int r = 0; r < 8; ++r) {
            const int mo = tm * 16 + r + hl * 8;
            out[((size_t)b * CO + mo) * NTOK + n0 + l15] = acc[r] + bc[mo];
        }
    }
}

// ---------------------------------------------------------------------------
// pixel_shuffle(s=2) + flatten: o2 (256,128,256) -> h (256,32768)
// ---------------------------------------------------------------------------
__global__ void shuffle_flatten(const float* __restrict__ o2, float* __restrict__ h)
{
    int idx = blockIdx.x * blockDim.x + threadIdx.x;
    if (idx >= BATCH * 128 * NTOK) return;
    int t  = idx & 255;
    int ch = (idx >> 8) & 127;
    int b  = idx >> 15;
    int hh = t >> 4, ww = t & 15;
    int c = ch >> 2, i = (ch >> 1) & 1, j = ch & 1;
    h[(size_t)b * 32768 + c * 1024 + (hh * 2 + i) * 32 + (ww * 2 + j)] = o2[idx];
}

// ---------------------------------------------------------------------------
extern "C" void kernel_launch(void* const* d_in, const int* in_sizes, int n_in,
                              void* d_out, int out_size, void* d_ws, size_t ws_size,
                              hipStream_t stream)
{
    (void)in_sizes; (void)n_in; (void)out_size; (void)ws_size;
    const float* x    = (const float*)d_in[0];
    const float* wq1  = (const float*)d_in[1];
    const float* bq1  = (const float*)d_in[2];
    const float* wk1  = (const float*)d_in[3];
    const float* bk1  = (const float*)d_in[4];
    const float* wv1  = (const float*)d_in[5];
    const float* bv1  = (const float*)d_in[6];
    const float* wc1  = (const float*)d_in[7];
    const float* bc1  = (const float*)d_in[8];
    const float* wo1  = (const float*)d_in[9];
    const float* bo1  = (const float*)d_in[10];
    const float* wq2  = (const float*)d_in[11];
    const float* bq2  = (const float*)d_in[12];
    const float* wk2  = (const float*)d_in[13];
    const float* bk2  = (const float*)d_in[14];
    const float* wv2  = (const float*)d_in[15];
    const float* bv2  = (const float*)d_in[16];
    const float* wc2  = (const float*)d_in[17];
    const float* bc2  = (const float*)d_in[18];
    const float* wo2  = (const float*)d_in[19];
    const float* bo2  = (const float*)d_in[20];
    const float* fc1w = (const float*)d_in[21];
    const float* fc1b = (const float*)d_in[22];
    const float* fc2w = (const float*)d_in[23];
    const float* fc2b = (const float*)d_in[24];
    float* outp = (float*)d_out;

    char* base = (char*)d_ws;
    size_t off = 0;
    auto wsa = [&](size_t bytes) -> void* {
        void* p = base + off;
        off = (off + bytes + 255) & ~(size_t)255;
        return p;
    };
    float* xu1  = (float*)wsa((size_t)256 * 12 * 256 * 4);
    float* q    = (float*)wsa((size_t)256 * 64 * 256 * 4);
    float* kbuf = (float*)wsa((size_t)256 * 64 * 256 * 4);
    float* vbuf = (float*)wsa((size_t)256 * 64 * 256 * 4);
    float* attn = (float*)wsa((size_t)256 * 256 * 9 * 4);
    int*   topi = (int*)  wsa((size_t)256 * 256 * 9 * 4);
    float* conv = (float*)wsa((size_t)256 * 128 * 256 * 4);
    float* lay  = (float*)wsa((size_t)256 * 128 * 256 * 4);
    float* hbuf = (float*)wsa((size_t)256 * 32768 * 4);
    float* fc1o = (float*)wsa((size_t)256 * 1024 * 4);

    // ---- layer 1 (shuffle after layer1 cancels with unshuffle before layer2) ----
    unshuffle12<<<(256 * 12 * 256 + 255) / 256, 256, 0, stream>>>(x, xu1);

    dim3 g1(256 / 32, (256 * 12) / 32);
    gemm_at_bt<false, false><<<g1, 32, 0, stream>>>(xu1, wq1, bq1, q,    256 * 12, 256, 256);
    gemm_at_bt<false, false><<<g1, 32, 0, stream>>>(xu1, wk1, bk1, kbuf, 256 * 12, 256, 256);
    gemm_at_bt<false, false><<<g1, 32, 0, stream>>>(xu1, wv1, bv1, vbuf, 256 * 12, 256, 256);

    topk_attn<12><<<(256 * 256) / 128, 128, 0, stream>>>(q, kbuf, attn, topi);
    conv_gather_wmma<12, 64><<<dim3(16, 256), 128, 0, stream>>>(vbuf, attn, topi, wc1, bc1, conv);
    gemm_at_bt<true, false><<<dim3(8, (256 * 64) / 32), 32, 0, stream>>>(conv, wo1, bo1, lay,
                                                                         256 * 64, 256, 256);

    // ---- layer 2 (input = lay, already in unshuffled (B,64,256) token layout) ----
    dim3 g2(256 / 32, (256 * 64) / 32);
    gemm_at_bt<false, false><<<g2, 32, 0, stream>>>(lay, wq2, bq2, q,    256 * 64, 256, 256);
    gemm_at_bt<false, false><<<g2, 32, 0, stream>>>(lay, wk2, bk2, kbuf, 256 * 64, 256, 256);
    gemm_at_bt<false, false><<<g2, 32, 0, stream>>>(lay, wv2, bv2, vbuf, 256 * 64, 256, 256);

    topk_attn<64><<<(256 * 256) / 128, 128, 0, stream>>>(q, kbuf, attn, topi);
    conv_gather_wmma<64, 128><<<dim3(16, 256), 128, 0, stream>>>(vbuf, attn, topi, wc2, bc2, conv);
    gemm_at_bt<true, false><<<dim3(8, (256 * 128) / 32), 32, 0, stream>>>(conv, wo2, bo2, lay,
                                                                          256 * 128, 256, 256);

    // ---- head ----
    shuffle_flatten<<<(256 * 128 * 256 + 255) / 256, 256, 0, stream>>>(lay, hbuf);
    gemm_at_bt<true , false><<<dim3(1024 / 32, 256 / 32), 32, 0, stream>>>(hbuf, fc1w, fc1b, fc1o,
                                                                           256, 1024, 32768);
    gemm_at_bt<false, true ><<<dim3(1, 256 / 32), 32, 0, stream>>>(fc1o, fc2w, fc2b, outp,
                                                                   256, 10, 1024);
}